// ElectricalGridModel_11768210391595
// MI455X (gfx1250) — compile-verified
//
#include <hip/hip_runtime.h>
#include <hip/hip_bf16.h>

// GAT (2x GATConv + Linear) for MI455X / gfx1250.
// Memory-bound edge phase (~2-3 GB random traffic, ~100us roofline @23.3TB/s,
// node arrays fit in 192MB L2). Dense GEMMs done with V_WMMA_F32_16X16X4_F32;
// A-operand staged through LDS so the k-loop is branch-free (ds_load + wmma).

typedef __attribute__((ext_vector_type(2))) float v2f;
typedef __attribute__((ext_vector_type(8))) float v8f;

#define LRELU_SLOPE 0.2f
#define HIDC 32

// ---- order-preserving float<->uint key for atomicMax-based segment max ----
__device__ __forceinline__ unsigned f32_key(float f) {
    unsigned u = __float_as_uint(f);
    return (u & 0x80000000u) ? ~u : (u | 0x80000000u);
}
__device__ __forceinline__ float f32_unkey(unsigned k) {
    unsigned u = (k & 0x80000000u) ? (k ^ 0x80000000u) : ~k;
    return __uint_as_float(u);
}

// ---------------------------------------------------------------- scalars --
__global__ void k_zero_scalars(float* scal) {
    if (blockIdx.x == 0 && threadIdx.x < 8) scal[threadIdx.x] = 0.f;
}

__global__ void k_sum_edge_attr(const float* __restrict__ ea, float* __restrict__ accum, int E) {
    __shared__ float sdata[256];
    float v = 0.f;
    for (long i = (long)blockIdx.x * blockDim.x + threadIdx.x; i < E; i += (long)gridDim.x * blockDim.x)
        v += ea[i];
    sdata[threadIdx.x] = v;
    __syncthreads();
    for (int s = 128; s > 0; s >>= 1) {
        if ((int)threadIdx.x < s) sdata[threadIdx.x] += sdata[threadIdx.x + s];
        __syncthreads();
    }
    if (threadIdx.x == 0) unsafeAtomicAdd(accum, sdata[0]);
}

// scal[0]=sum -> scal[1]=mean, scal[2]=ce1, scal[3]=ce2  (EDIM==1 so edge
// features enter alpha only through the scalar ce = dot(We, ae)).
__global__ void k_compute_scalars(float* scal, const float* We1, const float* ae1,
                                  const float* We2, const float* ae2, int E) {
    if (blockIdx.x || threadIdx.x) return;
    scal[1] = scal[0] / (float)E;
    float c1 = 0.f, c2 = 0.f;
    for (int j = 0; j < HIDC; ++j) { c1 += We1[j] * ae1[j]; c2 += We2[j] * ae2[j]; }
    scal[2] = c1; scal[3] = c2;
}

// ------------------------------------------------------------- WMMA GEMM --
// Y[N x Nout] = (reluX ? relu(X) : X)[N x Kin] @ W[Kin x Nout] (+ bias)
// 256 threads = 8 waves; each wave computes a 16-row x 32-col tile via two
// 16x16 f32 accumulators, stepping K by 4 with V_WMMA_F32_16X16X4_F32.
// X tile and W are staged in LDS (zero-padded, ReLU folded at staging time)
// so the inner loop is branch-free. X tile stride Kpad+1 avoids bank conflicts.
__global__ void k_wmma_gemm(const float* __restrict__ X, const float* __restrict__ W,
                            const float* __restrict__ bias, float* __restrict__ Y,
                            int N, int Kin, int Nout, int reluX) {
    __shared__ float Wl[32 * 32];         // [Kpad x 32], zero padded
    __shared__ float Xs[128 * 33];        // [128 rows x (Kpad+1)], zero padded
    const int tid  = threadIdx.x;
    const int Kpad = (Kin + 3) & ~3;      // 16 (layer1) or 32
    const int ldx  = Kpad + 1;            // LDS row stride (bank-conflict free)

    for (int idx = tid; idx < Kpad * 32; idx += 256) {
        int k = idx >> 5, n = idx & 31;
        Wl[idx] = (k < Kin && n < Nout) ? W[k * Nout + n] : 0.f;
    }

    const long rowBase = (long)blockIdx.x * 128;
    for (int idx = tid; idx < 128 * Kpad; idx += 256) {
        int r = idx / Kpad, k = idx - r * Kpad;
        long row = rowBase + r;
        float v = (row < N && k < Kin) ? X[row * Kin + k] : 0.f;
        if (reluX) v = fmaxf(v, 0.f);
        Xs[r * ldx + k] = v;
    }
    __syncthreads();

    const int wave  = tid >> 5;
    const int lane  = tid & 31;
    const int lrow  = lane & 15;   // M for A / N for B,C
    const int khalf = lane >> 4;   // 0 -> K{0,1}, 1 -> K{2,3} of the 4-wide step
    const float* xw = Xs + (wave * 16 + lrow) * ldx;   // this lane's A row

    v8f c0 = {}; v8f c1 = {};
    for (int kk = 0; kk < Kpad; kk += 4) {
        const int k0 = kk + khalf * 2;
        v2f a;  a.x  = xw[k0];                 a.y  = xw[k0 + 1];
        v2f b0; b0.x = Wl[k0 * 32 + lrow];     b0.y = Wl[(k0 + 1) * 32 + lrow];
        v2f b1; b1.x = Wl[k0 * 32 + 16 + lrow]; b1.y = Wl[(k0 + 1) * 32 + 16 + lrow];
        c0 = __builtin_amdgcn_wmma_f32_16x16x4_f32(false, a, false, b0, (short)0, c0, false, false);
        c1 = __builtin_amdgcn_wmma_f32_16x16x4_f32(false, a, false, b1, (short)0, c1, false, false);
    }

    // C/D layout: VGPR r, lanes 0-15 -> M=r ; lanes 16-31 -> M=8+r ; N=lane%16.
    const int mbase = khalf * 8;
    const long tileRow = rowBase + wave * 16;
    for (int r = 0; r < 8; ++r) {
        long row = tileRow + mbase + r;
        if (row < N) {
            int n0 = lrow;
            if (n0 < Nout) Y[row * Nout + n0] = c0[r] + (bias ? bias[n0] : 0.f);
            int n1 = 16 + lrow;
            if (n1 < Nout) Y[row * Nout + n1] = c1[r] + (bias ? bias[n1] : 0.f);
        }
    }
}

// ----------------------------------------------------- per-node attn dots --
// a_s[n] = h[n,:].as ; a_d[n] = h[n,:].ad   (one wave32 per node, shfl reduce)
__global__ void k_attn_scores(const float* __restrict__ H, const float* __restrict__ av_s,
                              const float* __restrict__ av_d, float* __restrict__ out_s,
                              float* __restrict__ out_d, int N) {
    long node = ((long)blockIdx.x * blockDim.x + threadIdx.x) >> 5;
    int  lane = threadIdx.x & 31;
    if (node >= N) return;
    float h  = H[node * HIDC + lane];
    float vs = h * av_s[lane];
    float vd = h * av_d[lane];
    for (int m = 16; m; m >>= 1) {
        vs += __shfl_xor(vs, m, 32);
        vd += __shfl_xor(vd, m, 32);
    }
    if (lane == 0) { out_s[node] = vs; out_d[node] = vd; }
}

// --------------------------------------------------------- per-layer init --
__global__ void k_init_layer(float* __restrict__ Out, const float* __restrict__ b,
                             unsigned* __restrict__ segmax, float* __restrict__ segsum, int N) {
    long i = (long)blockIdx.x * blockDim.x + threadIdx.x;
    if (i < (long)N * HIDC) Out[i] = b[i & (HIDC - 1)];
    if (i < N) { segmax[i] = 0u; segsum[i] = 0.f; }
}

// ----------------------------------------------- pass 1: alpha + seg max --
__global__ void k_edge_alpha_max(const int* __restrict__ ei, const float* __restrict__ eattr,
                                 const float* __restrict__ a_s, const float* __restrict__ a_d,
                                 const float* __restrict__ scal, int ce_idx,
                                 float* __restrict__ alpha, unsigned* __restrict__ segmax,
                                 int E, int N) {
    long e = (long)blockIdx.x * blockDim.x + threadIdx.x;
    long Etot = (long)E + N;
    if (e >= Etot) return;
    int s, d; float ea;
    if (e < E) { s = ei[e]; d = ei[(long)E + e]; ea = eattr[e]; }
    else       { s = d = (int)(e - E);           ea = scal[1]; }   // self loop, mean attr
    float a = a_s[s] + a_d[d] + ea * scal[ce_idx];
    a = (a > 0.f) ? a : LRELU_SLOPE * a;                            // leaky relu
    alpha[e] = a;
    atomicMax(&segmax[d], f32_key(a));
}

// ----------------------------------------------- pass 2: exp + seg sum ----
__global__ void k_edge_exp_sum(const int* __restrict__ ei, float* __restrict__ alpha,
                               const unsigned* __restrict__ segmax, float* __restrict__ segsum,
                               int E, int N) {
    long e = (long)blockIdx.x * blockDim.x + threadIdx.x;
    long Etot = (long)E + N;
    if (e >= Etot) return;
    int d = (e < E) ? ei[(long)E + e] : (int)(e - E);
    float w = expf(alpha[e] - f32_unkey(segmax[d]));
    alpha[e] = w;
    unsafeAtomicAdd(&segsum[d], w);
}

// -------------------------------- pass 3: weighted gather + scatter-add ---
// One wave32 per edge: lane j handles feature j. h[src] gather and out[dst]
// atomic-add are both 128B coalesced per wave -> GLOBAL_ATOMIC_ADD_F32.
__global__ void k_edge_aggregate(const int* __restrict__ ei, const float* __restrict__ alpha,
                                 const float* __restrict__ segsum, const float* __restrict__ H,
                                 float* __restrict__ Out, int E, int N) {
    long wid  = ((long)blockIdx.x * blockDim.x + threadIdx.x) >> 5;
    int  lane = threadIdx.x & 31;
    long Etot = (long)E + N;
    if (wid >= Etot) return;
    int s, d;
    if (wid < E) { s = ei[wid]; d = ei[(long)E + wid]; }
    else         { s = d = (int)(wid - E); }
    float w = alpha[wid] / (segsum[d] + 1e-16f);
    float v = H[(long)s * HIDC + lane] * w;
    unsafeAtomicAdd(&Out[(long)d * HIDC + lane], v);
}

// --------------------------------------------------------------- launch ---
extern "C" void kernel_launch(void* const* d_in, const int* in_sizes, int n_in,
                              void* d_out, int out_size, void* d_ws, size_t ws_size,
                              hipStream_t stream) {
    const float* x   = (const float*)d_in[0];
    const int*   ei  = (const int*)  d_in[1];   // [2,E]: row0 = src, row1 = dst
    const float* ea  = (const float*)d_in[2];
    const float* W1  = (const float*)d_in[3];
    const float* as1 = (const float*)d_in[4];
    const float* ad1 = (const float*)d_in[5];
    const float* We1 = (const float*)d_in[6];
    const float* ae1 = (const float*)d_in[7];
    const float* b1  = (const float*)d_in[8];
    const float* W2  = (const float*)d_in[9];
    const float* as2 = (const float*)d_in[10];
    const float* ad2 = (const float*)d_in[11];
    const float* We2 = (const float*)d_in[12];
    const float* ae2 = (const float*)d_in[13];
    const float* b2  = (const float*)d_in[14];
    const float* Wl  = (const float*)d_in[15];
    const float* bl  = (const float*)d_in[16];

    const int FIN = 14, OUTC = 20;
    const int N = in_sizes[0] / FIN;
    const int E = in_sizes[1] / 2;
    const long Etot = (long)E + N;

    // workspace carve-up (all re-initialized every launch)
    float* ws = (float*)d_ws;
    size_t off = 0;
    float*    h      = ws + off; off += (size_t)N * HIDC;
    float*    out1   = ws + off; off += (size_t)N * HIDC;
    float*    out2   = ws + off; off += (size_t)N * HIDC;
    float*    asv    = ws + off; off += N;
    float*    adv    = ws + off; off += N;
    float*    segsum = ws + off; off += N;
    unsigned* segmax = (unsigned*)(ws + off); off += N;
    float*    alpha  = ws + off; off += (size_t)Etot;
    float*    scal   = ws + off; off += 8;

    const int gemmBlocks  = (N + 127) / 128;
    const int nodeBlocks  = (N + 7) / 8;               // wave per node
    const int edgeTBlocks = (int)((Etot + 255) / 256); // thread per edge
    const int edgeWBlocks = (int)((Etot + 7) / 8);     // wave per edge
    const int initBlocks  = (int)(((long)N * HIDC + 255) / 256);

    // scalar prep
    k_zero_scalars<<<1, 8, 0, stream>>>(scal);
    k_sum_edge_attr<<<512, 256, 0, stream>>>(ea, scal, E);
    k_compute_scalars<<<1, 1, 0, stream>>>(scal, We1, ae1, We2, ae2, E);

    // ---- layer 1 ----
    k_wmma_gemm<<<gemmBlocks, 256, 0, stream>>>(x, W1, nullptr, h, N, FIN, HIDC, 0);
    k_attn_scores<<<nodeBlocks, 256, 0, stream>>>(h, as1, ad1, asv, adv, N);
    k_init_layer<<<initBlocks, 256, 0, stream>>>(out1, b1, segmax, segsum, N);
    k_edge_alpha_max<<<edgeTBlocks, 256, 0, stream>>>(ei, ea, asv, adv, scal, 2, alpha, segmax, E, N);
    k_edge_exp_sum<<<edgeTBlocks, 256, 0, stream>>>(ei, alpha, segmax, segsum, E, N);
    k_edge_aggregate<<<edgeWBlocks, 256, 0, stream>>>(ei, alpha, segsum, h, out1, E, N);

    // ---- layer 2 (ReLU on out1 folded into GEMM staging loads) ----
    k_wmma_gemm<<<gemmBlocks, 256, 0, stream>>>(out1, W2, nullptr, h, N, HIDC, HIDC, 1);
    k_attn_scores<<<nodeBlocks, 256, 0, stream>>>(h, as2, ad2, asv, adv, N);
    k_init_layer<<<initBlocks, 256, 0, stream>>>(out2, b2, segmax, segsum, N);
    k_edge_alpha_max<<<edgeTBlocks, 256, 0, stream>>>(ei, ea, asv, adv, scal, 3, alpha, segmax, E, N);
    k_edge_exp_sum<<<edgeTBlocks, 256, 0, stream>>>(ei, alpha, segmax, segsum, E, N);
    k_edge_aggregate<<<edgeWBlocks, 256, 0, stream>>>(ei, alpha, segsum, h, out2, E, N);

    // ---- final linear: d_out = out2 @ Wl + bl  (N x 20) ----
    k_wmma_gemm<<<gemmBlocks, 256, 0, stream>>>(out2, Wl, bl, (float*)d_out, N, HIDC, OUTC, 0);
}